// SmallNet_88252987998940
// MI455X (gfx1250) — compile-verified
//
#include <hip/hip_runtime.h>

// ---------------------------------------------------------------------------
// Hawkes-style likelihood:
//   out = E*beta - sum_e ||diff(u_e,v_e,t_e)||
//         - NON_EVENT_W * dx * sum_{p,s} exp(beta - ||diff(u_p,v_p,t_mid_s)||)
// Two grid-stride partial-sum kernels + one 1-wave finalize kernel.
// Wave reductions use V_WMMA_F32_16X16X4_F32 (A = lane partials, B = ones):
//   D[m][n] = x_m + x_{m+16}; summing the 8 accumulator VGPRs per lane and one
//   shfl_xor(16) yields the exact full wave32 sum with a fixed (deterministic)
//   dataflow. No float atomics anywhere -> bitwise reproducible across replays.
// ---------------------------------------------------------------------------

typedef float v2f __attribute__((ext_vector_type(2)));
typedef float v8f __attribute__((ext_vector_type(8)));

#define K_EPS 1e-6f
#define K_NSAMP 50
#define K_NON_EVENT_W 1.0f
#define NBLK 512
#define NTHR 256

// Full wave32 sum via one f32 WMMA. Requires EXEC == all ones (call only at
// uniform control-flow points with a full wave).
__device__ __forceinline__ float wave_sum32_wmma(float x) {
    v2f a; a[0] = x;    a[1] = 0.0f;   // A[16x4]: lane L<16 -> A[L][0..1], L>=16 -> A[L-16][2..3]
    v2f b; b[0] = 1.0f; b[1] = 1.0f;   // B[4x16] = all ones
    v8f c = {};
    c = __builtin_amdgcn_wmma_f32_16x16x4_f32(
            /*neg_a=*/false, a, /*neg_b=*/false, b,
            /*c_mod=*/(short)0, c, /*reuse_a=*/false, /*reuse_b=*/false);
    // c[r] @ lane n<16 = D[r][n], @ lane n>=16 = D[r+8][n-16]; D[m][*] = x_m + x_{m+16}
    float s = c[0] + c[1] + c[2] + c[3] + c[4] + c[5] + c[6] + c[7];
    s += __shfl_xor(s, 16, 32);        // (rows 0..7) + (rows 8..15)
    return s;                          // full 32-lane sum in every lane
}

// 256-thread block sum; result valid in wave 0 (all lanes).
__device__ __forceinline__ float block_sum256(float x) {
    __shared__ float wpart[NTHR / 32];
    const int lane = threadIdx.x & 31;
    const int wid  = threadIdx.x >> 5;
    float w = wave_sum32_wmma(x);
    if (lane == 0) wpart[wid] = w;
    __syncthreads();
    float r = 0.0f;
    if (wid == 0) {
        float v = (lane < (NTHR / 32)) ? wpart[lane] : 0.0f;
        r = wave_sum32_wmma(v);        // wave 0 fully active -> EXEC all ones
    }
    return r;
}

// ---- Kernel 1: sum of event distances ------------------------------------
__global__ void event_partial_kernel(const float2* __restrict__ z0,
                                     const float2* __restrict__ v0,
                                     const float2* __restrict__ a0,
                                     const float*  __restrict__ et,
                                     const int2*   __restrict__ euv,
                                     int E, float* __restrict__ part) {
    float acc = 0.0f;
    const int stride = gridDim.x * blockDim.x;
    for (int i = blockIdx.x * blockDim.x + threadIdx.x; i < E; i += stride) {
        const int2  uv = euv[i];
        const float t  = et[i];
        const float2 zu = z0[uv.x], zv = z0[uv.y];
        const float2 vu = v0[uv.x], vv = v0[uv.y];
        const float2 au = a0[uv.x], av = a0[uv.y];
        const float ht2 = 0.5f * t * t;
        const float dx = (zu.x - zv.x) + (vu.x - vv.x) * t + (au.x - av.x) * ht2 + K_EPS;
        const float dy = (zu.y - zv.y) + (vu.y - vv.y) * t + (au.y - av.y) * ht2 + K_EPS;
        acc += sqrtf(dx * dx + dy * dy);
    }
    const float b = block_sum256(acc);
    if (threadIdx.x == 0) part[blockIdx.x] = b;
}

// ---- Kernel 2: quadrature sum of exp(beta - d) over pairs x midpoints ----
__global__ void nonevent_partial_kernel(const float2* __restrict__ z0,
                                        const float2* __restrict__ v0,
                                        const float2* __restrict__ a0,
                                        const int*   __restrict__ pu,
                                        const int*   __restrict__ pv,
                                        int P,
                                        const float* __restrict__ beta,
                                        const int*   __restrict__ t0p,
                                        const int*   __restrict__ tnp,
                                        float* __restrict__ part) {
    const float bval = beta[0];
    const float t0f  = (float)t0p[0];
    const float tnf  = (float)tnp[0];
    const float dxq  = (tnf - t0f) / (float)K_NSAMP;
    float acc = 0.0f;
    const int stride = gridDim.x * blockDim.x;
    for (int i = blockIdx.x * blockDim.x + threadIdx.x; i < P; i += stride) {
        const int u = pu[i], v = pv[i];
        const float2 zu = z0[u], zv = z0[v];
        const float2 vu = v0[u], vv = v0[v];
        const float2 au = a0[u], av = a0[v];
        const float dzx = zu.x - zv.x, dzy = zu.y - zv.y;
        const float dvx = vu.x - vv.x, dvy = vu.y - vv.y;
        const float dax = 0.5f * (au.x - av.x), day = 0.5f * (au.y - av.y);
#pragma unroll 5
        for (int s = 0; s < K_NSAMP; ++s) {
            const float t  = t0f + ((float)s + 0.5f) * dxq;
            const float ex = dzx + dvx * t + dax * t * t + K_EPS;
            const float ey = dzy + dvy * t + day * t * t + K_EPS;
            const float dp = sqrtf(ex * ex + ey * ey);
            acc += __expf(bval - dp);   // v_exp_f32 path
        }
    }
    const float b = block_sum256(acc);
    if (threadIdx.x == 0) part[blockIdx.x] = b;
}

// ---- Kernel 3: single-wave finalize --------------------------------------
__global__ void finalize_kernel(const float* __restrict__ partE,
                                const float* __restrict__ partP,
                                int nE, int nP,
                                const float* __restrict__ beta,
                                const int*   __restrict__ t0p,
                                const int*   __restrict__ tnp,
                                float n_events,
                                float* __restrict__ out) {
    float se = 0.0f, sp = 0.0f;
    for (int i = threadIdx.x; i < nE; i += 32) se += partE[i];
    for (int i = threadIdx.x; i < nP; i += 32) sp += partP[i];
    se = wave_sum32_wmma(se);
    sp = wave_sum32_wmma(sp);
    if (threadIdx.x == 0) {
        const float t0f = (float)t0p[0];
        const float tnf = (float)tnp[0];
        const float dxq = (tnf - t0f) / (float)K_NSAMP;
        const float event_intensity = n_events * beta[0] - se;
        out[0] = event_intensity - K_NON_EVENT_W * sp * dxq;
    }
}

extern "C" void kernel_launch(void* const* d_in, const int* in_sizes, int n_in,
                              void* d_out, int out_size, void* d_ws, size_t ws_size,
                              hipStream_t stream) {
    (void)n_in; (void)out_size; (void)ws_size;
    const float*  beta = (const float*) d_in[0];
    const float2* z0   = (const float2*)d_in[1];
    const float2* v0   = (const float2*)d_in[2];
    const float2* a0   = (const float2*)d_in[3];
    const float*  et   = (const float*) d_in[4];
    const int2*   euv  = (const int2*)  d_in[5];
    const int*    pu   = (const int*)   d_in[6];
    const int*    pv   = (const int*)   d_in[7];
    const int*    t0p  = (const int*)   d_in[8];
    const int*    tnp  = (const int*)   d_in[9];

    const int E = in_sizes[4];      // 1,000,000 events
    const int P = in_sizes[6];      // 200,000 pairs

    float* partE = (float*)d_ws;            // NBLK floats
    float* partP = ((float*)d_ws) + NBLK;   // NBLK floats

    event_partial_kernel<<<NBLK, NTHR, 0, stream>>>(z0, v0, a0, et, euv, E, partE);
    nonevent_partial_kernel<<<NBLK, NTHR, 0, stream>>>(z0, v0, a0, pu, pv, P,
                                                       beta, t0p, tnp, partP);
    finalize_kernel<<<1, 32, 0, stream>>>(partE, partP, NBLK, NBLK,
                                          beta, t0p, tnp, (float)E, (float*)d_out);
}